// Distance_PBC_18081812316191
// MI455X (gfx1250) — compile-verified
//
#include <hip/hip_runtime.h>

// ---------------------------------------------------------------------------
// PBC radius graph (27-image, cutoff 6.0, strict 32-nearest) for MI455X.
// d2[i,c] = |p_i|^2 + |q_c|^2 - 2 p_i . q_c computed as 16x16 tiles with
// V_WMMA_F32_16X16X4_F32 (A=[x,y,z,1], B=[-2qx,-2qy,-2qz,|q|^2], C=|p_i|^2).
// pos[] staged to LDS via GLOBAL_LOAD_ASYNC_TO_LDS_B128 (ASYNCcnt-tracked,
// drained with s_wait_asynccnt). Candidate compaction + rank-based top-K
// selection in LDS.
// ---------------------------------------------------------------------------

#define N_ATOMS  2048
#define K_NBR    32
#define N_IMG    27
#define ZERO_OFF 13
#define CUTOFF2  36.0f
#define CAND     (N_IMG * N_ATOMS)     // 55296
#define CTILES   (CAND / 16)           // 3456
#define ROWS_PB  16                    // atoms per block (one WMMA row tile)
#define WAVES_PB 8
#define BLOCK    (WAVES_PB * 32)       // 256
#define TILES_PW (CTILES / WAVES_PB)   // 432
#define CAP      192                   // per-row candidate buffer capacity
#define N_EDGE   (N_ATOMS * K_NBR + N_ATOMS)  // 67584

typedef __attribute__((ext_vector_type(2))) float v2f;
typedef __attribute__((ext_vector_type(8))) float v8f;
typedef int v4i __attribute__((vector_size(16)));   // builtin pointee type

#define AS1 __attribute__((address_space(1)))
#define AS3 __attribute__((address_space(3)))

#if __has_builtin(__builtin_amdgcn_global_load_async_to_lds_b128)
#define ASYNC_LDS_B128 1
#elif __has_builtin(__builtin_amdgcn_global_load_async_to_lds_b32)
#define ASYNC_LDS_B32 1
#endif

__device__ __forceinline__ void wait_asynccnt0() {
#if __has_builtin(__builtin_amdgcn_s_wait_asynccnt)
    __builtin_amdgcn_s_wait_asynccnt(0);
#else
    asm volatile("s_wait_asynccnt 0" ::: "memory");
#endif
}

__global__ __launch_bounds__(BLOCK)
void pbc_graph_kernel(const float* __restrict__ pos,
                      const float* __restrict__ cell,
                      int*   __restrict__ out_idx,   // [2, N_EDGE] row-major
                      float* __restrict__ out_w,     // [N_EDGE]
                      float* __restrict__ out_vec)   // [N_EDGE, 3]
{
    __shared__ float    s_pos[N_ATOMS * 3];   // 24 KB: all atom positions
    __shared__ float    s_off[N_IMG * 3];     // 27 cartesian image shifts
    __shared__ float    s_d2[ROWS_PB][CAP];   // 12 KB
    __shared__ unsigned s_id[ROWS_PB][CAP];   // 12 KB
    __shared__ unsigned s_cnt[ROWS_PB];
    __shared__ float    s_norm[ROWS_PB];

    const int t    = threadIdx.x;
    const int lane = t & 31;
    const int wave = t >> 5;
    const int hl   = lane >> 4;      // lane half (0: lanes 0-15, 1: lanes 16-31)
    const int m    = lane & 15;      // row (A) / column (B,D) within tile

    // ---- stage positions into LDS (async-to-LDS when available) -----------
#if defined(ASYNC_LDS_B128)
    {
        // 2048*3*4 B = 24576 B = 1536 16-byte chunks; 256 lanes x 6 chunks.
        AS1 v4i* gsrc = (AS1 v4i*)(uintptr_t)pos;
        AS3 v4i* ldst = (AS3 v4i*)(unsigned)(uintptr_t)&s_pos[0];
        #pragma unroll
        for (int k = 0; k < 6; ++k) {
            const int e = t + k * BLOCK;     // 16-byte chunk index
            __builtin_amdgcn_global_load_async_to_lds_b128(gsrc + e, ldst + e,
                                                           0 /*offset*/, 0 /*cpol*/);
        }
        wait_asynccnt0();
    }
#elif defined(ASYNC_LDS_B32)
    {
        AS1 int* gsrc = (AS1 int*)(uintptr_t)pos;
        AS3 int* ldst = (AS3 int*)(unsigned)(uintptr_t)&s_pos[0];
        #pragma unroll
        for (int k = 0; k < 24; ++k) {
            const int e = t + k * BLOCK;     // dword index
            __builtin_amdgcn_global_load_async_to_lds_b32(gsrc + e, ldst + e,
                                                          0 /*offset*/, 0 /*cpol*/);
        }
        wait_asynccnt0();
    }
#else
    for (int a = t; a < N_ATOMS * 3; a += BLOCK) s_pos[a] = pos[a];
#endif
    if (t < N_IMG) {
        const float fx = (float)(t / 9 - 1);
        const float fy = (float)((t / 3) % 3 - 1);
        const float fz = (float)(t % 3 - 1);
        #pragma unroll
        for (int d = 0; d < 3; ++d)
            s_off[t * 3 + d] = fx * cell[0 + d] + fy * cell[3 + d] + fz * cell[6 + d];
    }
    if (t < ROWS_PB) s_cnt[t] = 0u;
    __syncthreads();

    const int i0 = blockIdx.x * ROWS_PB;     // first target atom of this block

    // ---- build per-wave constant A operand and C (=|p_i|^2) operand -------
    const int   im = i0 + m;
    const float px = s_pos[im * 3 + 0];
    const float py = s_pos[im * 3 + 1];
    const float pz = s_pos[im * 3 + 2];
    if (wave == 0 && lane < 16) s_norm[m] = px * px + py * py + pz * pz;
    __syncthreads();

    // A (16x4 f32): lanes 0-15 hold K={0,1}; lanes 16-31 hold K={2,3}
    v2f av;
    av.x = hl ? pz : px;
    av.y = hl ? 1.0f : py;
    // C (16x16 f32): VGPR r <-> row (r + 8*hl); same value for every column
    v8f cnorm;
    #pragma unroll
    for (int r = 0; r < 8; ++r) cnorm[r] = s_norm[r + 8 * hl];

    // ---- scan: 16x16 d2 tiles via V_WMMA_F32_16X16X4_F32 -------------------
    const int ct_beg = wave * TILES_PW;
    const int ct_end = ct_beg + TILES_PW;
    #pragma unroll 2
    for (int ct = ct_beg; ct < ct_end; ++ct) {
        const int c = ct * 16 + m;            // this lane's candidate column
        const int o = c >> 11;                // image index  (c / 2048)
        const int j = c & (N_ATOMS - 1);      // source atom  (c % 2048)
        const float qx = s_pos[j * 3 + 0] + s_off[o * 3 + 0];
        const float qy = s_pos[j * 3 + 1] + s_off[o * 3 + 1];
        const float qz = s_pos[j * 3 + 2] + s_off[o * 3 + 2];

        // B (4x16 f32): lanes 0-15 hold K={0,1}; lanes 16-31 hold K={2,3}
        v2f bv;
        bv.x = hl ? (-2.0f * qz) : (-2.0f * qx);
        bv.y = hl ? (qx * qx + qy * qy + qz * qz) : (-2.0f * qy);

        // D = A*B + C  -> 16x16 tile of squared distances (uniform EXEC)
        v8f d = __builtin_amdgcn_wmma_f32_16x16x4_f32(
            false, av, false, bv, (short)0, cnorm, false, false);

        #pragma unroll
        for (int r = 0; r < 8; ++r) {
            const int   row  = r + 8 * hl;
            const int   ia   = i0 + row;
            const float d2   = d[r];
            const bool  self = (o == ZERO_OFF) && (j == ia);
            if (d2 <= CUTOFF2 && !self) {
                const unsigned slot = atomicAdd(&s_cnt[row], 1u); // ds_add_rtn_u32
                if (slot < CAP) {
                    s_d2[row][slot] = d2;
                    s_id[row][slot] = (unsigned)c;
                }
            }
        }
    }
    __syncthreads();

    // ---- rank-based top-K selection (matches jax top_k tie-breaking) -------
    const int rows_pw = ROWS_PB / WAVES_PB;   // 2 rows per wave
    for (int rr = 0; rr < rows_pw; ++rr) {
        const int row = wave * rows_pw + rr;
        const int ia  = i0 + row;
        unsigned  cnt = s_cnt[row];
        const int mc  = (cnt < (unsigned)CAP) ? (int)cnt : CAP;
        const float pix = s_pos[ia * 3 + 0];
        const float piy = s_pos[ia * 3 + 1];
        const float piz = s_pos[ia * 3 + 2];

        for (int e = lane; e < mc; e += 32) {
            const float    de = s_d2[row][e];
            const unsigned ie = s_id[row][e];
            int rank = 0;
            for (int f = 0; f < mc; ++f) {
                const float    df = s_d2[row][f];
                const unsigned jf = s_id[row][f];
                rank += (df < de) || (df == de && jf < ie);
            }
            if (rank < K_NBR) {
                const int o = (int)(ie >> 11);
                const int j = (int)(ie & (N_ATOMS - 1));
                const float vx = s_pos[j * 3 + 0] + s_off[o * 3 + 0] - pix;
                const float vy = s_pos[j * 3 + 1] + s_off[o * 3 + 1] - piy;
                const float vz = s_pos[j * 3 + 2] + s_off[o * 3 + 2] - piz;
                const float w  = sqrtf(vx * vx + vy * vy + vz * vz);
                const int ep = ia * K_NBR + rank;
                out_idx[ep]          = j;    // edge_index[0] (source)
                out_idx[N_EDGE + ep] = ia;   // edge_index[1] (target)
                out_w[ep]            = w;
                out_vec[ep * 3 + 0]  = vx;
                out_vec[ep * 3 + 1]  = vy;
                out_vec[ep * 3 + 2]  = vz;
            }
        }
        // pad slots [mc, K) as zero-weight self edges (reference semantics)
        for (int slot = mc + lane; slot < K_NBR; slot += 32) {
            const int ep = ia * K_NBR + slot;
            out_idx[ep]          = ia;
            out_idx[N_EDGE + ep] = ia;
            out_w[ep]            = 0.0f;
            out_vec[ep * 3 + 0]  = 0.0f;
            out_vec[ep * 3 + 1]  = 0.0f;
            out_vec[ep * 3 + 2]  = 0.0f;
        }
    }

    // ---- appended self loops -----------------------------------------------
    if (t < ROWS_PB) {
        const int ia = i0 + t;
        const int ep = N_ATOMS * K_NBR + ia;
        out_idx[ep]          = ia;
        out_idx[N_EDGE + ep] = ia;
        out_w[ep]            = 0.0f;
        out_vec[ep * 3 + 0]  = 0.0f;
        out_vec[ep * 3 + 1]  = 0.0f;
        out_vec[ep * 3 + 2]  = 0.0f;
    }
}

extern "C" void kernel_launch(void* const* d_in, const int* in_sizes, int n_in,
                              void* d_out, int out_size, void* d_ws, size_t ws_size,
                              hipStream_t stream) {
    (void)in_sizes; (void)n_in; (void)out_size; (void)d_ws; (void)ws_size;
    const float* pos  = (const float*)d_in[0];   // [2048,3] f32
    const float* cell = (const float*)d_in[1];   // [3,3]    f32

    // d_out layout (flat, return order):
    //   [0            , 2*N_EDGE) : edge_index  (int32, [2, N_EDGE])
    //   [2*N_EDGE     , 3*N_EDGE) : edge_weight (f32)
    //   [3*N_EDGE     , 6*N_EDGE) : edge_vec    (f32, [N_EDGE, 3])
    int*   out_idx = (int*)d_out;
    float* out_w   = (float*)d_out + 2 * N_EDGE;
    float* out_vec = (float*)d_out + 3 * N_EDGE;

    pbc_graph_kernel<<<N_ATOMS / ROWS_PB, BLOCK, 0, stream>>>(
        pos, cell, out_idx, out_w, out_vec);
}